// GraphTRSS_26147760898595
// MI455X (gfx1250) — compile-verified
//
#include <hip/hip_runtime.h>

typedef float v2f __attribute__((ext_vector_type(2)));
typedef float v8f __attribute__((ext_vector_type(8)));

#define Bb 32
#define Tt 256
#define Nn 128
#define Dd 64

// ---------------------------------------------------------------------------
// Phase 1: per-batch build M = L + 1.1*I and invert in place (Gauss-Jordan,
// no pivoting -- M is strictly diagonally dominant: diag = sum_{j!=i} A_ij + 1.1).
// 32 blocks x 512 threads; 4 threads per matrix row (32 columns each).
// LDS: 128 x 129 (padded) matrix + 512 partial-sum slots (dynamic, ~68 KB).
// ---------------------------------------------------------------------------
__global__ __launch_bounds__(512) void graph_inv_kernel(
    const float* __restrict__ A, float* __restrict__ P) {
  extern __shared__ float sm[];           // [128][129] + degp[512]
  float* degp = sm + 128 * 129;

  const int b   = blockIdx.x;
  const int tid = threadIdx.x;            // 0..511
  const int row = tid >> 2;               // 0..127
  const int q   = tid & 3;                // column quarter
  const int j0  = q * 32;

  const float* Ab = A + (size_t)b * Nn * Nn;

  // M = -A; accumulate row degree
  float s = 0.0f;
  #pragma unroll 8
  for (int j = j0; j < j0 + 32; ++j) {
    float v = Ab[row * Nn + j];
    sm[row * 129 + j] = -v;
    s += v;
  }
  degp[tid] = s;
  __syncthreads();
  if (q == 0) {
    float d = degp[row * 4 + 0] + degp[row * 4 + 1] +
              degp[row * 4 + 2] + degp[row * 4 + 3];
    // diag of M = (deg_i - A_ii) + 1.1  (Laplacian diag + 1.1)
    sm[row * 129 + row] += d + 1.1f;
  }
  __syncthreads();

  // In-place Gauss-Jordan inversion
  for (int k = 0; k < Nn; ++k) {
    float ip = 1.0f / sm[k * 129 + k];
    __syncthreads();                      // everyone has read the pivot
    if (row == k) {
      #pragma unroll 8
      for (int j = j0; j < j0 + 32; ++j)
        sm[k * 129 + j] = (j == k) ? ip : sm[k * 129 + j] * ip;
    }
    __syncthreads();                      // pivot row final
    float f = sm[row * 129 + k];
    __syncthreads();                      // col-k reads done before writes
    if (row != k) {
      #pragma unroll 8
      for (int j = j0; j < j0 + 32; ++j) {
        float rk = sm[k * 129 + j];
        sm[row * 129 + j] = (j == k) ? (-f * ip) : (sm[row * 129 + j] - f * rk);
      }
    }
    __syncthreads();                      // iteration complete
  }

  // Write P (dense 128x128 per batch) to workspace
  float* Pb = P + (size_t)b * Nn * Nn;
  for (int idx = tid; idx < Nn * Nn; idx += 512)
    Pb[idx] = sm[(idx >> 7) * 129 + (idx & 127)];
}

// ---------------------------------------------------------------------------
// Phase 2: scan  y_t = y_{t-1} + P (x_t - y_{t-1})
// Grid: 128 blocks = 32 batches x 4 column-slices (16 cols each).
// Block: 256 threads = 8 waves; wave w owns output rows [16w, 16w+16).
// y kept in WMMA f32 C-layout accumulators; P rows held in 64 VGPRs/lane
// (loaded once, reused 256 steps -- no LDS for A).
// LDS: ping-pong u[2][128][16], stride 16 so each B-operand pair
// (K rows k and k+1) sits exactly 64 B apart -> single ds_load_2addr_b32
// straight into the WMMA src1 register pair (no mov reassembly, partial
// dscnt waits).  B reads are bank-conflict-free (lanes 16-31 offset +32
// dwords); only the 8 u-writes/lane see a 2-way conflict (cheap).
// One barrier per step (ping-pong makes the trailing barrier unnecessary).
// ---------------------------------------------------------------------------
#define USTRIDE 16

__global__ __launch_bounds__(256) void graph_scan_kernel(
    const float* __restrict__ xs, const float* __restrict__ P,
    float* __restrict__ out) {
  __shared__ float sU[2][Nn * USTRIDE];   // 16 KB

  const int b    = blockIdx.x >> 2;
  const int d0   = (blockIdx.x & 3) * 16;
  const int tid  = threadIdx.x;
  const int wave = tid >> 5;               // 0..7
  const int lane = tid & 31;
  const int n    = lane & 15;              // output column within slice
  const int hi   = lane >> 4;              // 0/1 : row-half / K-pair select
  const int rowBase = wave * 16;

  // A operands in registers: row (rowBase+n) of P, K pair {k0+2hi, k0+2hi+1}
  // for k0 = 0,4,...,124.  (16x4 f32 A-layout: lane->M, vgpr v -> K = v + 2*hi)
  const float* Prow =
      P + (size_t)b * Nn * Nn + (size_t)(rowBase + n) * Nn + 2 * hi;
  v2f aReg[32];
  #pragma unroll
  for (int i = 0; i < 32; ++i)
    aReg[i] = *(const v2f*)&Prow[i * 4];

  const float* xb = xs  + ((size_t)b * Tt) * (Nn * Dd) + d0;
  float*       yb = out + ((size_t)b * Tt) * (Nn * Dd) + d0;

  v8f acc = {0.f, 0.f, 0.f, 0.f, 0.f, 0.f, 0.f, 0.f};   // y_0 = 0

  for (int t = 0; t < Tt; ++t) {
    float* ub = sU[t & 1];
    const float* xt = xb + (size_t)t * (Nn * Dd);

    // u = x_t - y_{t-1}  (each wave writes its 16 rows of the 128x16 tile)
    #pragma unroll
    for (int r = 0; r < 8; ++r) {
      int row = rowBase + r + 8 * hi;      // matches f32 C-layout
      ub[row * USTRIDE + n] = xt[row * Dd + n] - acc[r];
    }
    __syncthreads();

    // acc = y_{t-1} + P[rowBase:rowBase+16, :] @ u   via f32 WMMA, K=128
    #pragma unroll
    for (int i = 0; i < 32; ++i) {
      int k0 = i * 4;
      // B tile 4x16: vgpr v -> K = k0 + v + 2*hi, lane -> N = n
      // consecutive K rows are +USTRIDE dwords -> one ds_load_2addr_b32
      v2f bv;
      bv.x = ub[(k0 + 2 * hi) * USTRIDE + n];
      bv.y = ub[(k0 + 2 * hi + 1) * USTRIDE + n];
      acc = __builtin_amdgcn_wmma_f32_16x16x4_f32(
          /*neg_a=*/false, aReg[i], /*neg_b=*/false, bv,
          /*c_mod=*/(short)0, acc, /*reuse_a=*/false, /*reuse_b=*/false);
    }

    // store y_t
    float* yt = yb + (size_t)t * (Nn * Dd);
    #pragma unroll
    for (int r = 0; r < 8; ++r) {
      int row = rowBase + r + 8 * hi;
      yt[row * Dd + n] = acc[r];
    }
    // no trailing barrier: next step writes the OTHER u buffer; the earlier
    // barrier plus dscnt waits before this step's WMMAs order buffer reuse.
  }
}

// ---------------------------------------------------------------------------
extern "C" void kernel_launch(void* const* d_in, const int* in_sizes, int n_in,
                              void* d_out, int out_size, void* d_ws, size_t ws_size,
                              hipStream_t stream) {
  const float* xs = (const float*)d_in[0];   // [32,256,128,64]
  const float* A  = (const float*)d_in[1];   // [32,128,128]
  float* out = (float*)d_out;                // [32,256,128,64]
  float* P   = (float*)d_ws;                 // 32*128*128 floats = 2 MB scratch

  const int sh_inv = (128 * 129 + 512) * sizeof(float);   // ~68 KB dynamic LDS

  // >64KB dynamic LDS for the inversion kernel (host-side, graph-capture safe)
  (void)hipFuncSetAttribute((const void*)graph_inv_kernel,
                            hipFuncAttributeMaxDynamicSharedMemorySize, sh_inv);

  graph_inv_kernel<<<Bb, 512, sh_inv, stream>>>(A, P);
  graph_scan_kernel<<<Bb * 4, 256, 0, stream>>>(xs, P, out);
}